// GNNModel_19679540150705
// MI455X (gfx1250) — compile-verified
//
#include <hip/hip_runtime.h>
#include <hip/hip_bf16.h>

typedef float v2f __attribute__((ext_vector_type(2)));
typedef float v8f __attribute__((ext_vector_type(8)));

#define MIN_NORM 1e-15f
#define SQRT_C   1e-3f          // sqrt(1e-6)
#define MAXNORM  996.0f         // (1-0.004)/sqrt(c)

__device__ __forceinline__ float tanh_clip(float t) {
    return tanhf(fminf(fmaxf(t, -15.f), 15.f));
}
__device__ __forceinline__ float artanh_clip(float x) {
    x = fminf(fmaxf(x, -1.f + 1e-5f), 1.f - 1e-5f);
    return 0.5f * (log1pf(x) - log1pf(-x));
}

// ---------------------------------------------------------------------------
// Zero fill
// ---------------------------------------------------------------------------
__global__ void zero_f32(float* __restrict__ p, long long n) {
    long long i = (long long)blockIdx.x * blockDim.x + threadIdx.x;
    if (i < n) p[i] = 0.f;
}

// ---------------------------------------------------------------------------
// f32 WMMA 16x16x4 fragment layouts (wave32):
// A (16x4): lane(g=lane>>4, m=lane&15), vgpr v <-> A[m][2g+v]
// B (4x16): lane(g,m),                  vgpr v <-> B[2g+v][m] = W[m][2g+v]
// C/D (16x16): lane(g,m),               vgpr v <-> D[8g+v][m]
// ---------------------------------------------------------------------------

// M=16 tile variant, specialized at compile time so the K-loop is branch-free.
template <bool GATHER, bool TAIL, bool BIAS>
__global__ void gemm64_m16(const float* __restrict__ X,
                           const float* __restrict__ W,
                           const float* __restrict__ bias,
                           const int*   __restrict__ rowidx,
                           float* __restrict__ Y,
                           int nrows)
{
    const int lane = threadIdx.x & 31;
    const int wave = threadIdx.x >> 5;
    const int wavesPerBlock = blockDim.x >> 5;
    const int ntiles = (nrows + 15) >> 4;
    const int tile = blockIdx.x * wavesPerBlock + wave;
    if (tile >= ntiles) return;

    const int row0 = tile << 4;
    const int m = lane & 15;
    const int g = lane >> 4;

    // Hoist gather + bounds handling OUT of the K loop.
    int r = row0 + m;
    float amask = 1.f;
    if (TAIL && r >= nrows) { r = nrows - 1; amask = 0.f; }
    const int ar = GATHER ? rowidx[r] : r;
    const float* __restrict__ Arow = X + (size_t)ar * 64 + g * 2;
    const float* __restrict__ Brow = W + (size_t)m * 64 + g * 2;

    v8f acc[4] = {};

    #pragma unroll
    for (int kb = 0; kb < 64; kb += 4) {
        // Batch all loads for this K-step so they issue as one clause,
        // then run the WMMA burst behind a single wait.
        v2f a = *(const v2f*)(Arow + kb);
        v2f b[4];
        #pragma unroll
        for (int nt = 0; nt < 4; ++nt)
            b[nt] = *(const v2f*)(Brow + (size_t)nt * 16 * 64 + kb);
        if (TAIL) { a.x *= amask; a.y *= amask; }
        #pragma unroll
        for (int nt = 0; nt < 4; ++nt)
            acc[nt] = __builtin_amdgcn_wmma_f32_16x16x4_f32(
                false, a, false, b[nt], (short)0, acc[nt], false, false);
    }

    #pragma unroll
    for (int nt = 0; nt < 4; ++nt) {
        const float bval = BIAS ? bias[nt * 16 + m] : 0.f;
        #pragma unroll
        for (int v = 0; v < 8; ++v) {
            const int row = row0 + g * 8 + v;
            if (!TAIL || row < nrows)
                Y[(size_t)row * 64 + nt * 16 + m] = acc[nt][v] + bval;
        }
    }
}

// M=32 double-tile variant (nrows % 32 == 0, no gather/bias/tail):
// two row tiles share each B fragment -> 8 WMMAs per 6 b64 loads,
// with all 6 loads batched ahead of the WMMA burst.
__global__ void gemm64_m32(const float* __restrict__ X,
                           const float* __restrict__ W,
                           float* __restrict__ Y,
                           int nrows)
{
    const int lane = threadIdx.x & 31;
    const int wave = threadIdx.x >> 5;
    const int wavesPerBlock = blockDim.x >> 5;
    const int npairs = nrows >> 5;
    const int pair = blockIdx.x * wavesPerBlock + wave;
    if (pair >= npairs) return;

    const int row0 = pair << 5;
    const int m = lane & 15;
    const int g = lane >> 4;

    const float* __restrict__ Arow0 = X + (size_t)(row0 + m) * 64 + g * 2;
    const float* __restrict__ Arow1 = Arow0 + 16 * 64;
    const float* __restrict__ Brow  = W + (size_t)m * 64 + g * 2;

    v8f acc[8] = {};

    #pragma unroll
    for (int kb = 0; kb < 64; kb += 4) {
        v2f a0 = *(const v2f*)(Arow0 + kb);
        v2f a1 = *(const v2f*)(Arow1 + kb);
        v2f b[4];
        #pragma unroll
        for (int nt = 0; nt < 4; ++nt)
            b[nt] = *(const v2f*)(Brow + (size_t)nt * 16 * 64 + kb);
        #pragma unroll
        for (int nt = 0; nt < 4; ++nt) {
            acc[nt]     = __builtin_amdgcn_wmma_f32_16x16x4_f32(
                false, a0, false, b[nt], (short)0, acc[nt], false, false);
            acc[nt + 4] = __builtin_amdgcn_wmma_f32_16x16x4_f32(
                false, a1, false, b[nt], (short)0, acc[nt + 4], false, false);
        }
    }

    #pragma unroll
    for (int nt = 0; nt < 4; ++nt) {
        #pragma unroll
        for (int v = 0; v < 8; ++v) {
            const int row = row0 + g * 8 + v;
            Y[(size_t)row * 64 + nt * 16 + m]        = acc[nt][v];
            Y[(size_t)(row + 16) * 64 + nt * 16 + m] = acc[nt + 4][v];
        }
    }
}

// ---------------------------------------------------------------------------
// Per-edge kernel: one wave per edge, lane l owns components 2l,2l+1.
// pre = WsH[sub] + WrH[rel] + WqrHb[r_idx]; alpha = sigmoid(relu(pre)·Wattn)
// mess2 = P*hs + Q*hr with scalars P,Q from {||hs||^2, ||hr||^2, hs·hr}
// atomic scatter-add of alpha*mess2 into agg[obj].
// ---------------------------------------------------------------------------
__global__ void edge_kernel(const int*   __restrict__ edges,
                            const float* __restrict__ hidden,
                            const float* __restrict__ rela,
                            const float* __restrict__ WsH,
                            const float* __restrict__ WrH,
                            const float* __restrict__ WqrHb,
                            const float* __restrict__ Wattn,
                            float* __restrict__ agg,
                            int n_edge)
{
    const int e    = (int)((blockIdx.x * (long long)blockDim.x + threadIdx.x) >> 5);
    const int lane = threadIdx.x & 31;
    if (e >= n_edge) return;

    const int* ep = edges + (size_t)e * 6;
    const int r_idx = ep[0];
    const int rel   = ep[2];
    const int sub   = ep[4];
    const int obj   = ep[5];

    const v2f hs = *(const v2f*)(hidden + (size_t)sub * 64 + 2 * lane);
    const v2f hr = *(const v2f*)(rela   + (size_t)rel * 64 + 2 * lane);
    const v2f p1 = *(const v2f*)(WsH    + (size_t)sub   * 64 + 2 * lane);
    const v2f p2 = *(const v2f*)(WrH    + (size_t)rel   * 64 + 2 * lane);
    const v2f p3 = *(const v2f*)(WqrHb  + (size_t)r_idx * 64 + 2 * lane);
    const v2f wa = *(const v2f*)(Wattn  + 2 * lane);

    const float prex = p1.x + p2.x + p3.x;
    const float prey = p1.y + p2.y + p3.y;

    float s_at = fmaxf(prex, 0.f) * wa.x + fmaxf(prey, 0.f) * wa.y;
    float s_ss = hs.x * hs.x + hs.y * hs.y;
    float s_rr = hr.x * hr.x + hr.y * hr.y;
    float s_sr = hs.x * hr.x + hs.y * hr.y;

    #pragma unroll
    for (int off = 16; off > 0; off >>= 1) {
        s_at += __shfl_xor(s_at, off, 32);
        s_ss += __shfl_xor(s_ss, off, 32);
        s_rr += __shfl_xor(s_rr, off, 32);
        s_sr += __shfl_xor(s_sr, off, 32);
    }

    const float alpha = 1.f / (1.f + expf(-s_at));

    const float c = 1e-6f;
    // expmap0 + project scaling for hs and hr
    float n_s = fmaxf(sqrtf(s_ss), MIN_NORM);
    float ks  = tanh_clip(SQRT_C * n_s) / (SQRT_C * n_s);
    float nsh = ks * n_s;
    if (nsh > MAXNORM) ks *= MAXNORM / nsh;

    float n_r = fmaxf(sqrtf(s_rr), MIN_NORM);
    float kr  = tanh_clip(SQRT_C * n_r) / (SQRT_C * n_r);
    float nrh = kr * n_r;
    if (nrh > MAXNORM) kr *= MAXNORM / nrh;

    // mobius_add(ks*hs, kr*hr, c) -> P0*hs + Q0*hr
    const float x2 = ks * ks * s_ss;
    const float y2 = kr * kr * s_rr;
    const float xy = ks * kr * s_sr;
    const float A  = 1.f + 2.f * c * xy + c * y2;
    const float Bc = 1.f - c * x2;
    const float den = fmaxf(1.f + 2.f * c * xy + c * c * x2 * y2, MIN_NORM);
    float P0 = A * ks / den;
    float Q0 = Bc * kr / den;

    // project(m) then logmap0
    float nm2 = P0 * P0 * s_ss + 2.f * P0 * Q0 * s_sr + Q0 * Q0 * s_rr;
    float nm  = fmaxf(sqrtf(fmaxf(nm2, 0.f)), MIN_NORM);
    if (nm > MAXNORM) { const float sc = MAXNORM / nm; P0 *= sc; Q0 *= sc; nm = MAXNORM; }
    const float f = artanh_clip(SQRT_C * nm) / (nm * SQRT_C);

    const float P = alpha * f * P0;
    const float Q = alpha * f * Q0;

    float* dst = agg + (size_t)obj * 64 + 2 * lane;
    unsafeAtomicAdd(dst + 0, P * hs.x + Q * hr.x);
    unsafeAtomicAdd(dst + 1, P * hs.y + Q * hr.y);
}

// ---------------------------------------------------------------------------
// In-place finalize on Y (nnode x 64): logmap0(relu(expmap0(Y)))
// relu(k*v) = k*relu(v) since k > 0, so only two norms are needed per row.
// ---------------------------------------------------------------------------
__global__ void finalize_kernel(float* __restrict__ Y, int nrows)
{
    const int row  = (int)((blockIdx.x * (long long)blockDim.x + threadIdx.x) >> 5);
    const int lane = threadIdx.x & 31;
    if (row >= nrows) return;

    v2f v = *(const v2f*)(Y + (size_t)row * 64 + 2 * lane);
    const float rx = fmaxf(v.x, 0.f);
    const float ry = fmaxf(v.y, 0.f);

    float s_vv = v.x * v.x + v.y * v.y;
    float s_rr = rx * rx + ry * ry;
    #pragma unroll
    for (int off = 16; off > 0; off >>= 1) {
        s_vv += __shfl_xor(s_vv, off, 32);
        s_rr += __shfl_xor(s_rr, off, 32);
    }

    float n_v = fmaxf(sqrtf(s_vv), MIN_NORM);
    float k   = tanh_clip(SQRT_C * n_v) / (SQRT_C * n_v);
    float nh  = k * n_v;
    if (nh > MAXNORM) k *= MAXNORM / nh;

    const float n_a = fmaxf(k * sqrtf(s_rr), MIN_NORM);
    const float f   = artanh_clip(SQRT_C * n_a) / (n_a * SQRT_C);

    v2f out;
    out.x = f * k * rx;
    out.y = f * k * ry;
    *(v2f*)(Y + (size_t)row * 64 + 2 * lane) = out;
}

// ---------------------------------------------------------------------------
// Host-side dispatch helper for plain (no gather, no bias) GEMMs.
// ---------------------------------------------------------------------------
static inline void launch_gemm_plain(const float* X, const float* W, float* Y,
                                     int nrows, hipStream_t stream)
{
    if ((nrows & 31) == 0) {
        int pairs = nrows >> 5;
        gemm64_m32<<<(pairs + 3) / 4, 128, 0, stream>>>(X, W, Y, nrows);
    } else if ((nrows & 15) == 0) {
        int tiles = nrows >> 4;
        gemm64_m16<false, false, false><<<(tiles + 3) / 4, 128, 0, stream>>>(
            X, W, nullptr, nullptr, Y, nrows);
    } else {
        int tiles = (nrows + 15) >> 4;
        gemm64_m16<false, true, false><<<(tiles + 3) / 4, 128, 0, stream>>>(
            X, W, nullptr, nullptr, Y, nrows);
    }
}

extern "C" void kernel_launch(void* const* d_in, const int* in_sizes, int n_in,
                              void* d_out, int out_size, void* d_ws, size_t ws_size,
                              hipStream_t stream)
{
    const float* hidden = (const float*)d_in[0];
    const int*   q_rel  = (const int*)  d_in[1];
    const int*   edges  = (const int*)  d_in[2];
    const float* rela   = (const float*)d_in[3];
    const float* Ws_w   = (const float*)d_in[4];
    const float* Wr_w   = (const float*)d_in[5];
    const float* Wqr_w  = (const float*)d_in[6];
    const float* Wqr_b  = (const float*)d_in[7];
    const float* Wattn  = (const float*)d_in[8];
    const float* Wh_w   = (const float*)d_in[9];

    const int nnode  = in_sizes[0] / 64;
    const int nbatch = in_sizes[1];
    const int nedge  = in_sizes[2] / 6;
    const int nvocab = in_sizes[3] / 64;

    float* WsH   = (float*)d_ws;                       // nnode  x 64
    float* agg   = WsH   + (size_t)nnode  * 64;        // nnode  x 64
    float* WrH   = agg   + (size_t)nnode  * 64;        // nvocab x 64
    float* WqrHb = WrH   + (size_t)nvocab * 64;        // nbatch x 64
    float* outF  = (float*)d_out;                      // nnode  x 64

    // 1) zero agg
    {
        long long n = (long long)nnode * 64;
        zero_f32<<<(unsigned)((n + 255) / 256), 256, 0, stream>>>(agg, n);
    }
    // 2) WsH = hidden @ Ws^T
    launch_gemm_plain(hidden, Ws_w, WsH, nnode, stream);
    // 3) WrH = rela @ Wr^T  (401 rows -> tail variant)
    {
        int tiles = (nvocab + 15) >> 4;
        if ((nvocab & 15) == 0)
            gemm64_m16<false, false, false><<<(tiles + 3) / 4, 128, 0, stream>>>(
                rela, Wr_w, nullptr, nullptr, WrH, nvocab);
        else
            gemm64_m16<false, true, false><<<(tiles + 3) / 4, 128, 0, stream>>>(
                rela, Wr_w, nullptr, nullptr, WrH, nvocab);
    }
    // 4) WqrHb = rela[q_rel] @ Wqr^T + b  (gather + bias)
    {
        int tiles = (nbatch + 15) >> 4;
        if ((nbatch & 15) == 0)
            gemm64_m16<true, false, true><<<(tiles + 3) / 4, 128, 0, stream>>>(
                rela, Wqr_w, Wqr_b, q_rel, WqrHb, nbatch);
        else
            gemm64_m16<true, true, true><<<(tiles + 3) / 4, 128, 0, stream>>>(
                rela, Wqr_w, Wqr_b, q_rel, WqrHb, nbatch);
    }
    // 5) per-edge message + scatter-add (one wave per edge)
    {
        long long blocks = ((long long)nedge * 32 + 255) / 256;
        edge_kernel<<<(unsigned)blocks, 256, 0, stream>>>(edges, hidden, rela, WsH, WrH, WqrHb,
                                                          Wattn, agg, nedge);
    }
    // 6) d_out = agg @ Wh^T (WMMA)
    launch_gemm_plain(agg, Wh_w, outF, nnode, stream);
    // 7) finalize in place
    {
        long long blocks = ((long long)nnode * 32 + 255) / 256;
        finalize_kernel<<<(unsigned)blocks, 256, 0, stream>>>(outF, nnode);
    }
}